// GCNConvCustom_57045755625628
// MI455X (gfx1250) — compile-verified
//
#include <hip/hip_runtime.h>

typedef float v2f __attribute__((ext_vector_type(2)));
typedef float v8f __attribute__((ext_vector_type(8)));

#define N_NODES 8192
#define N_EDGES 131072
#define D_FEAT 128
#define WPR 256          // bitmap words per row = 8192/32
#define GCN_EPS 1e-5f

// ---------------- 1. zero the adjacency bitmap (8 MB) ----------------
__global__ void gcn_zero_kernel(uint4* __restrict__ p, int n4) {
    int i = blockIdx.x * blockDim.x + threadIdx.x;
    if (i < n4) p[i] = uint4{0u, 0u, 0u, 0u};
}

// ---------------- 2. scatter edges into symmetric bitmap ----------------
__global__ void gcn_scatter_kernel(const int* __restrict__ ei,
                                   unsigned* __restrict__ bm) {
    int e = blockIdx.x * blockDim.x + threadIdx.x;
    if (e >= N_EDGES) return;
    int r = ei[e];
    int c = ei[N_EDGES + e];
    atomicOr(&bm[(size_t)r * WPR + (c >> 5)], 1u << (c & 31));
    atomicOr(&bm[(size_t)c * WPR + (r >> 5)], 1u << (r & 31));
}

// ---------------- 3. degree -> d^{-1/2} (one wave32 per row) ----------------
__global__ void gcn_degree_kernel(const unsigned* __restrict__ bm,
                                  float* __restrict__ dinv) {
    int wave = threadIdx.x >> 5;
    int lane = threadIdx.x & 31;
    int row  = blockIdx.x * 8 + wave;
    const unsigned* p = bm + (size_t)row * WPR;
    int s = 0;
#pragma unroll
    for (int t = 0; t < 8; ++t) s += __popc(p[lane + 32 * t]);
#pragma unroll
    for (int m = 16; m >= 1; m >>= 1) s += __shfl_xor(s, m, 32);
    if (lane == 0) {
        // row-sum of A = popcount(row) + 1 (added identity)
        dinv[row] = rsqrtf((float)s + 1.0f + GCN_EPS);
    }
}

// ---------------- 4. pre-scale source features: Xs = dinv[j] * X[j] ----------------
__global__ void gcn_scale_kernel(const float* __restrict__ x,
                                 const float* __restrict__ dinv,
                                 float* __restrict__ xs) {
    int i = blockIdx.x * blockDim.x + threadIdx.x;   // N*D threads
    xs[i] = x[i] * dinv[i >> 7];                     // D_FEAT == 128
}

// ---------------- 5. sparse aggregation: Y[i] = dinv[i]*(sum_{bit j} Xs[j] + Xs[i]) ----
__global__ void gcn_aggregate_kernel(const unsigned* __restrict__ bm,
                                     const float* __restrict__ xs,
                                     const float* __restrict__ dinv,
                                     float* __restrict__ y) {
    int row = blockIdx.x;
    int d   = threadIdx.x;                 // 128 threads: one per feature
    const unsigned* p = bm + (size_t)row * WPR;
    float acc = xs[(size_t)row * D_FEAT + d];   // identity (self-loop) term
    for (int w = 0; w < WPR; ++w) {
        unsigned bits = p[w];              // uniform across block -> scalar/broadcast
        int base = w << 5;
        while (bits) {
            int j = base + __ffs(bits) - 1;
            bits &= bits - 1;
            acc += xs[(size_t)j * D_FEAT + d];   // coalesced 512B row gather
        }
    }
    y[(size_t)row * D_FEAT + d] = acc * dinv[row];
}

// ---------------- 6. Out = Y @ W + bias via V_WMMA_F32_16X16X4_F32 ----------------
// 8 waves/block; wave w owns the 16x16 tile at (blockIdx.x*16, w*16).
__global__ void gcn_gemm_kernel(const float* __restrict__ y,
                                const float* __restrict__ w,
                                const float* __restrict__ bias,
                                float* __restrict__ out) {
    int wave = threadIdx.x >> 5;           // 0..7 -> column tile
    int lane = threadIdx.x & 31;
    int half = lane >> 4;                  // 0: K=k0,k0+1   1: K=k0+2,k0+3
    int l    = lane & 15;
    int rowBase = blockIdx.x * 16;
    int col     = wave * 16 + l;

    const float* arow = y + (size_t)(rowBase + l) * D_FEAT;
    v8f c = {};
#pragma unroll
    for (int k0 = 0; k0 < D_FEAT; k0 += 4) {
        int ka = k0 + half * 2;
        v2f a, b;
        a.x = arow[ka];
        a.y = arow[ka + 1];
        b.x = w[(size_t)ka * D_FEAT + col];
        b.y = w[(size_t)(ka + 1) * D_FEAT + col];
        c = __builtin_amdgcn_wmma_f32_16x16x4_f32(
                /*neg_a=*/false, a, /*neg_b=*/false, b,
                /*c_mod=*/(short)0, c, /*reuse_a=*/false, /*reuse_b=*/false);
    }
    float bv = bias[col];
#pragma unroll
    for (int v = 0; v < 8; ++v) {
        int r = rowBase + v + half * 8;    // C/D layout: VGPR v -> M=v (+8 for upper lanes)
        out[(size_t)r * D_FEAT + col] = c[v] + bv;
    }
}

extern "C" void kernel_launch(void* const* d_in, const int* in_sizes, int n_in,
                              void* d_out, int out_size, void* d_ws, size_t ws_size,
                              hipStream_t stream) {
    const float* x    = (const float*)d_in[0];
    const int*   ei   = (const int*)d_in[1];
    const float* wgt  = (const float*)d_in[2];
    const float* bias = (const float*)d_in[3];
    float* out = (float*)d_out;

    char* ws = (char*)d_ws;
    unsigned* bm  = (unsigned*)ws;                                  // 8 MB bitmap
    float* dinv   = (float*)(ws + (size_t)N_NODES * WPR * 4);       // 32 KB
    float* xs     = dinv + N_NODES;                                 // 4 MB
    float* y      = xs + (size_t)N_NODES * D_FEAT;                  // 4 MB

    int n4 = N_NODES * WPR / 4;
    gcn_zero_kernel<<<n4 / 256, 256, 0, stream>>>((uint4*)bm, n4);
    gcn_scatter_kernel<<<(N_EDGES + 255) / 256, 256, 0, stream>>>(ei, bm);
    gcn_degree_kernel<<<N_NODES / 8, 256, 0, stream>>>(bm, dinv);
    gcn_scale_kernel<<<N_NODES * D_FEAT / 256, 256, 0, stream>>>(x, dinv, xs);
    gcn_aggregate_kernel<<<N_NODES, 128, 0, stream>>>(bm, xs, dinv, y);
    gcn_gemm_kernel<<<N_NODES / 16, 256, 0, stream>>>(y, wgt, bias, out);
}